// PRNNLayer_3152505995892
// MI455X (gfx1250) — compile-verified
//
#include <hip/hip_runtime.h>
#include <stdint.h>

// ---------------------------------------------------------------------------
// GR4J-style scan: B=4096 independent basins, T=1024 sequential steps.
// Latency-bound (only 128 wave32s of parallelism). Strategy:
//   * 1 thread = 1 basin; block 256 = 8 waves/WGP (2 per SIMD32) for
//     latency interleaving; 16 workgroups.
//   * Double-buffered async global->LDS staging of 8-timestep chunks
//     (global_load_async_to_lds_b128 + s_wait_asynccnt) — each thread owns a
//     private padded LDS row, so NO barriers are required.
//   * Shortest dependent chain: v_exp_f32 / v_tanh_f32 / v_sqrt / v_rsq /
//     v_rcp; x^-0.25 computed as rsqrt(sqrt(x)); the gw term is exactly 0
//     (X2 == 0) so its powf(.,3.5) is elided.
// ---------------------------------------------------------------------------

#define BLOCK 256
#define TC 8                               // timesteps per chunk
#define ROWDW (TC * 3 + 4)                 // 28 dwords: 96B data + 16B pad (16B-aligned rows)
#define LOADS_PER_CHUNK ((TC * 3 * 4) / 16) // 6 x b128 per row per chunk

typedef int v4i __attribute__((ext_vector_type(4)));
typedef __attribute__((address_space(1))) v4i GV4;   // global int4
typedef __attribute__((address_space(3))) v4i LV4;   // LDS int4

__device__ __forceinline__ float frcp(float x) { return __builtin_amdgcn_rcpf(x); }

// x^(-1/4) = rsqrt(sqrt(x)), x > 0 here
__device__ __forceinline__ float pm025(float x) {
  return __builtin_amdgcn_rsqf(__builtin_amdgcn_sqrtf(x));
}

__device__ __forceinline__ float exp_fast(float x) {
  return __builtin_amdgcn_exp2f(x * 1.44269504089f);
}

// arguments are always >= 0 in this model
__device__ __forceinline__ float tanh_fast_nonneg(float x) {
#if __has_builtin(__builtin_amdgcn_tanhf)
  return __builtin_amdgcn_tanhf(x);            // native v_tanh_f32 on gfx1250
#else
  float e2x = __builtin_amdgcn_exp2f(x * 2.88539008178f); // e^(2x)
  return 1.0f - 2.0f * frcp(e2x + 1.0f);
#endif
}

// one 16B async DMA transfer: global -> LDS (ASYNCcnt)
__device__ __forceinline__ void async_b128(const void* g, uint32_t lds_off) {
#if __has_builtin(__builtin_amdgcn_global_load_async_to_lds_b128)
  __builtin_amdgcn_global_load_async_to_lds_b128((GV4*)(uintptr_t)g,
                                                 (LV4*)lds_off, 0, 0);
#else
  asm volatile("global_load_async_to_lds_b128 %0, %1, off"
               :: "v"(lds_off), "v"((unsigned long long)(uintptr_t)g)
               : "memory");
#endif
}

template <int N>
__device__ __forceinline__ void wait_asynccnt() {
#if __has_builtin(__builtin_amdgcn_s_wait_asynccnt)
  __builtin_amdgcn_s_wait_asynccnt(N);
#else
  asm volatile("s_wait_asynccnt %0" :: "i"(N) : "memory");
#endif
  asm volatile("" ::: "memory");  // keep LDS reads below the wait
}

__global__ __launch_bounds__(BLOCK) void gr4j_scan_kernel(
    const float* __restrict__ inp,    // (B, T, 3)
    const float* __restrict__ prod0,  // (B, 1)
    const float* __restrict__ rout0,  // (B, 1)
    float* __restrict__ out,          // B*T flow ++ B prod_f ++ B rout_f
    int Bn, int Tn) {
  __shared__ float lds[2 * BLOCK * ROWDW];  // double-buffered private rows

  const int tid = threadIdx.x;
  const int b = blockIdx.x * BLOCK + tid;

  const float* gbase = inp + (size_t)b * (size_t)Tn * 3u;
  float* flow = out + (size_t)b * (size_t)Tn;

  float prod_s = prod0[b];
  float rout_s = rout0[b];

  float* row0p = &lds[tid * ROWDW];
  float* row1p = &lds[BLOCK * ROWDW + tid * ROWDW];
  const uint32_t l0 = (uint32_t)(uintptr_t)row0p;  // low 32 bits = LDS byte addr
  const uint32_t l1 = (uint32_t)(uintptr_t)row1p;

  // prologue: stream chunk 0 into buffer 0
#pragma unroll
  for (int j = 0; j < LOADS_PER_CHUNK; ++j)
    async_b128((const char*)gbase + j * 16, l0 + j * 16);

  const int nchunks = Tn / TC;
  float qbuf[4];

  // model constants (X1=1000.5, X3=150.5, X4=15.5; X2==0 exactly)
  const float invX1 = 9.99500250e-4f;       // 1/1000.5
  const float invX3 = 6.64451827e-3f;       // 1/150.5
  const float UH1c = 9.51509538e-4f;        // 0.9 * (1/15.5)^2.5
  const float UH2c = 5.28616411e-5f;        // 0.1 * 0.5 * (1/15.5)^2.5

  for (int c = 0; c < nchunks; ++c) {
    const float* row = (c & 1) ? row1p : row0p;
    const uint32_t nxt = (c & 1) ? l0 : l1;

    if (c + 1 < nchunks) {
      const char* g = (const char*)gbase + (size_t)(c + 1) * (TC * 3 * 4);
#pragma unroll
      for (int j = 0; j < LOADS_PER_CHUNK; ++j)
        async_b128(g + j * 16, nxt + j * 16);
      // async loads complete in order: <=LOADS_PER_CHUNK outstanding means the
      // current chunk has fully landed while the next streams behind it.
      wait_asynccnt<LOADS_PER_CHUNK>();
    } else {
      wait_asynccnt<0>();
    }

#pragma unroll
    for (int j = 0; j < TC; ++j) {
      const float p  = row[j * 3 + 0];
      const float tm = row[j * 3 + 1];
      const float dl = row[j * 3 + 2];

      // PET: 29.8 * (dayl*24) * 0.611 * exp(17.3*tm/(tm+237.3)) / (tm+273.2)
      const float pet = 436.9872f * dl *
          exp_fast(17.3f * tm * frcp(tm + 237.3f)) * frcp(tm + 273.2f);

      const bool cond = p >= pet;
      const float p_n  = cond ? (p - pet) : 0.0f;
      const float pe_n = cond ? 0.0f : (pet - p);

      const float tn = tanh_fast_nonneg(p_n * invX1);
      const float te = tanh_fast_nonneg(pe_n * invX1);

      const float sr = prod_s * invX1;  // prod_s / X1
      const float p_s = cond
          ? (1000.5f * (1.0f - sr * sr) * tn) * frcp(1.0f + sr * tn)
          : 0.0f;
      const float e_s = cond
          ? 0.0f
          : (prod_s * (2.0f - sr) * te) * frcp(1.0f + (1.0f - sr) * te);

      const float s_new = prod_s - e_s + p_s;
      const float y  = 0.444444444f * s_new * invX1;  // (4/9)*s_new/X1
      const float y2 = y * y;
      const float perc = s_new * (1.0f - pm025(1.0f + y2 * y2));

      const float p_r = perc + (p_n - p_s);
      const float uh1 = UH1c * p_r;
      const float uh2 = UH2c * p_r;

      // gw = X2 * (...)^3.5 with X2 == 0  ->  gw == 0
      const float s_r = fmaxf(0.0f, rout_s + uh1);
      const float z  = s_r * invX3;
      const float z2 = z * z;
      const float q_r = s_r * (1.0f - pm025(1.0f + z2 * z2));
      const float q_d = fmaxf(0.0f, uh2);

      qbuf[j & 3] = q_r + q_d;
      if ((j & 3) == 3) {
        float4 v = make_float4(qbuf[0], qbuf[1], qbuf[2], qbuf[3]);
        *reinterpret_cast<float4*>(flow + c * TC + (j - 3)) = v;  // b128 store
      }

      prod_s += fminf(fmaxf(s_new - perc, -100000.0f), 100000.0f);
      rout_s += fminf(fmaxf(s_r - q_r,   -100000.0f), 100000.0f);
    }
  }

  out[(size_t)Bn * Tn + b] = prod_s;        // prod_f
  out[(size_t)Bn * Tn + Bn + b] = rout_s;   // rout_f
}

extern "C" void kernel_launch(void* const* d_in, const int* in_sizes, int n_in,
                              void* d_out, int out_size, void* d_ws, size_t ws_size,
                              hipStream_t stream) {
  (void)n_in; (void)out_size; (void)d_ws; (void)ws_size;
  const float* inp = (const float*)d_in[0];
  const float* p0  = (const float*)d_in[1];
  const float* r0  = (const float*)d_in[2];
  float* out = (float*)d_out;

  const int Bn = in_sizes[1];              // 4096 (prod_s0 flat count)
  const int Tn = in_sizes[0] / (Bn * 3);   // 1024

  dim3 grid(Bn / BLOCK), block(BLOCK);
  gr4j_scan_kernel<<<grid, block, 0, stream>>>(inp, p0, r0, out, Bn, Tn);
}